// NADE_34686155882588
// MI455X (gfx1250) — compile-verified
//
#include <hip/hip_runtime.h>
#include <stdint.h>

#define B_DIM 8192
#define D_DIM 1024
#define H_DIM 1024
#define NWAVE 16
#define HPW   (H_DIM / NWAVE)      // 64 h-rows per wave
#define BT    32                   // batch columns per workgroup (== wave32 lanes)
#define NTHREADS (NWAVE * 32)      // 512
#define CHUNK 32                   // d-steps of x staged per LDS refill
#define XPAD  34                   // padded row stride (dwords), conflict-free & 8B aligned

// ---- CDNA5 async global->LDS helpers (gfx1250, tracked with ASYNCcnt) ----
__device__ __forceinline__ unsigned lds_off(const void* p) {
    // generic pointers to LDS carry the workgroup-relative offset in the low 32 bits
    return (unsigned)(unsigned long long)p;
}

__device__ __forceinline__ void async_load_b64(unsigned lds, unsigned voff,
                                               const void* base) {
    // GVS mode: mem = SADDR(64) + VADDR(32); VDST = LDS byte address
    asm volatile("global_load_async_to_lds_b64 %0, %1, %2"
                 :
                 : "v"(lds), "v"(voff), "s"((unsigned long long)base)
                 : "memory");
}

__device__ __forceinline__ void wait_async0() {
    asm volatile("s_wait_asynccnt 0" ::: "memory");
}

// sigmoid with raw transcendentals: v_exp_f32 (exp2) + v_rcp_f32
__device__ __forceinline__ float fast_sigmoid(float x) {
    float e = __builtin_amdgcn_exp2f(x * -1.44269504088896340736f);
    return __builtin_amdgcn_rcpf(1.0f + e);
}

// ---------------- W (H x D) -> Wt (D x H) transpose pre-pass ----------------
__global__ __launch_bounds__(256) void nade_transpose_kernel(
    const float* __restrict__ src, float* __restrict__ dst) {
    __shared__ float tile[32][33];
    const int tx = threadIdx.x;           // 0..31
    const int ty = threadIdx.y;           // 0..7
    const int x0 = blockIdx.x * 32;       // d
    const int y0 = blockIdx.y * 32;       // h
#pragma unroll
    for (int j = 0; j < 32; j += 8)
        tile[ty + j][tx] = src[(size_t)(y0 + ty + j) * D_DIM + (x0 + tx)];
    __syncthreads();
#pragma unroll
    for (int j = 0; j < 32; j += 8)
        dst[(size_t)(x0 + ty + j) * H_DIM + (y0 + tx)] = tile[tx][ty + j];
}

// ------------------------------ main NADE scan ------------------------------
__global__ __launch_bounds__(NTHREADS) void nade_scan_kernel(
    const float* __restrict__ x,     // (B, D)
    const float* __restrict__ V,     // (D, H)
    const float* __restrict__ bias,  // (D)
    const float* __restrict__ Wt,    // (D, H)  == W^T, from workspace
    const float* __restrict__ c,     // (H)
    float* __restrict__ out)         // (B, D)
{
    __shared__ __align__(16) float s_vw[2][NWAVE][2 * HPW];  // [step-parity][wave][V|W]
    __shared__ __align__(16) float s_x[2][BT][XPAD];         // [chunk-parity] x tiles
    __shared__ __align__(16) float s_red[2][NWAVE][BT];      // [step-parity] partials

    const int tid = threadIdx.x;
    const int w   = tid >> 5;            // wave id 0..15
    const int l   = tid & 31;            // lane  -> batch column
    const int b0  = blockIdx.x * BT;
    const int h0  = w * HPW;

    // per-thread state a[h, b0+l] for h in [h0, h0+HPW)
    float a[HPW];
#pragma unroll
    for (int i = 0; i < HPW; ++i) a[i] = c[h0 + i];

    const int xr = tid >> 4;   // 0..31  : batch row within tile
    const int xg = tid & 15;   // 0..15  : 2-float group within 32-step chunk

    // prologue: async-stage V/W rows for d = 0 and x chunk 0
    {
        unsigned voff = (unsigned)((h0 + l * 2) * 4);       // d = 0
        async_load_b64(lds_off(&s_vw[0][w][l * 2]),       voff, V);
        async_load_b64(lds_off(&s_vw[0][w][HPW + l * 2]), voff, Wt);
        unsigned voffX = (unsigned)(((b0 + xr) * D_DIM + xg * 2) * 4);
        async_load_b64(lds_off(&s_x[0][xr][xg * 2]), voffX, x);
    }

    for (int d = 0; d < D_DIM; ++d) {
        const int buf = d & 1;              // step parity (V/W, reduction buffers)
        const int cp  = (d >> 5) & 1;       // chunk parity (x tile buffer)

        wait_async0();                      // V/W for step d (+ x tile at chunk head)
        if ((d & (CHUNK - 1)) == 0)
            __syncthreads();                // make all waves' async x writes visible

        // prefetch V/W rows for step d+1 into the other step buffer
        if (d + 1 < D_DIM) {
            unsigned voff = (unsigned)(((d + 1) * H_DIM + h0 + l * 2) * 4);
            async_load_b64(lds_off(&s_vw[buf ^ 1][w][l * 2]),       voff, V);
            async_load_b64(lds_off(&s_vw[buf ^ 1][w][HPW + l * 2]), voff, Wt);
        }
        // prefetch next x chunk one step early so its latency hides under compute
        if ((d & (CHUNK - 1)) == (CHUNK - 1) && d + 1 < D_DIM) {
            unsigned voffX =
                (unsigned)(((b0 + xr) * D_DIM + (d + 1) + xg * 2) * 4);
            async_load_b64(lds_off(&s_x[cp ^ 1][xr][xg * 2]), voffX, x);
        }

        const float xd = s_x[cp][l][d & (CHUNK - 1)];   // conflict-free (pad 34)
        const float4* vp = (const float4*)&s_vw[buf][w][0];
        const float4* wp = (const float4*)&s_vw[buf][w][HPW];

        float acc0 = 0.f, acc1 = 0.f, acc2 = 0.f, acc3 = 0.f;
#pragma unroll
        for (int i = 0; i < HPW / 4; ++i) {
            float4 v4 = vp[i];      // broadcast ds_load_b128 (uniform addr)
            float4 w4 = wp[i];
            acc0 = fmaf(fast_sigmoid(a[4 * i + 0]), v4.x, acc0);
            a[4 * i + 0] = fmaf(w4.x, xd, a[4 * i + 0]);
            acc1 = fmaf(fast_sigmoid(a[4 * i + 1]), v4.y, acc1);
            a[4 * i + 1] = fmaf(w4.y, xd, a[4 * i + 1]);
            acc2 = fmaf(fast_sigmoid(a[4 * i + 2]), v4.z, acc2);
            a[4 * i + 2] = fmaf(w4.z, xd, a[4 * i + 2]);
            acc3 = fmaf(fast_sigmoid(a[4 * i + 3]), v4.w, acc3);
            a[4 * i + 3] = fmaf(w4.w, xd, a[4 * i + 3]);
        }

        // cross-wave reduction over the 16 waves' h-partials (parity buffers)
        s_red[buf][w][l] = (acc0 + acc1) + (acc2 + acc3);
        __syncthreads();
        if (w == 0) {
            float s = bias[d];
#pragma unroll
            for (int ww = 0; ww < NWAVE; ++ww) s += s_red[buf][ww][l];
            out[(size_t)(b0 + l) * D_DIM + d] = fast_sigmoid(s);
        }
    }
}

extern "C" void kernel_launch(void* const* d_in, const int* in_sizes, int n_in,
                              void* d_out, int out_size, void* d_ws, size_t ws_size,
                              hipStream_t stream) {
    const float* x = (const float*)d_in[0];   // (B, D)
    const float* V = (const float*)d_in[1];   // (D, H)
    const float* b = (const float*)d_in[2];   // (D)
    const float* W = (const float*)d_in[3];   // (H, D)
    const float* c = (const float*)d_in[4];   // (H)
    float* out = (float*)d_out;               // (B, D)
    float* Wt  = (float*)d_ws;                // (D, H) scratch, 4 MB

    hipLaunchKernelGGL(nade_transpose_kernel,
                       dim3(D_DIM / 32, H_DIM / 32), dim3(32, 8), 0, stream,
                       W, Wt);

    hipLaunchKernelGGL(nade_scan_kernel,
                       dim3(B_DIM / BT), dim3(NTHREADS), 0, stream,
                       x, V, b, Wt, c, out);
}